// VariationalLinearEncoder_54924041781478
// MI455X (gfx1250) — compile-verified
//
#include <hip/hip_runtime.h>

#define N_NODES 100000
#define N_EDGES 1600000
#define D_IN    256
#define D_OUT   32
#define D2      64          // combined mu|logstd output width

typedef float v2f __attribute__((ext_vector_type(2)));
typedef float v8f __attribute__((ext_vector_type(8)));

// ---------------- degree / normalization ----------------

__global__ void k_deg_init(float* __restrict__ deg) {
    int i = blockIdx.x * blockDim.x + threadIdx.x;
    if (i < N_NODES) deg[i] = 1.0f;                 // self-loop contribution
}

__global__ void k_deg_count(const int* __restrict__ dst, float* __restrict__ deg) {
    int e = blockIdx.x * blockDim.x + threadIdx.x;
    if (e < N_EDGES) unsafeAtomicAdd(&deg[dst[e]], 1.0f);  // exact: deg < 2^24
}

__global__ void k_dinv(float* __restrict__ deg) {
    int i = blockIdx.x * blockDim.x + threadIdx.x;
    if (i < N_NODES) deg[i] = rsqrtf(deg[i]);       // deg >= 1 always
}

// ---------------- fused GEMM: h[N,64] = x[N,256] @ [W_mu | W_logstd] ----------------
// Block = 256 threads = 8 waves. Wave w: row tile = (w>>2)*16 within 32-row block,
// col tile = (w&3)*16 of the 64 combined cols. K loop: 64 x V_WMMA_F32_16X16X4_F32.
//
// ISA 32-bit operand striping (wave32):
//   A 16x4 : lane l<16 holds A[m=l][K=k+0 (v0), k+1 (v1)]; lane l>=16 holds K=k+2,k+3
//   B 4x16 : v0 = B[k+0][n] (lanes 0-15) / B[k+2][n] (lanes 16-31); v1 = K+1 / K+3
//   C/D 16x16 f32: vgpr j = row j (lanes 0-15) / row j+8 (lanes 16-31), col = lane&15

__global__ __launch_bounds__(256) void k_gemm(const float* __restrict__ x,
                                              const float* __restrict__ Wmu,
                                              const float* __restrict__ Wls,
                                              float* __restrict__ h) {
    const int lane  = threadIdx.x & 31;
    const int wave  = threadIdx.x >> 5;
    const int rbase = blockIdx.x * 32 + ((wave >> 2) << 4);  // 16-row tile base
    const int ct    = wave & 3;                              // 16-col tile 0..3
    const float* __restrict__ W = (ct < 2) ? Wmu : Wls;      // [256 x 32] row-major
    const int c0    = (ct & 1) << 4;                         // 0 or 16 inside W
    const int koff  = (lane >> 4) << 1;                      // 0 or 2
    const int m     = rbase + (lane & 15);
    const int n     = c0 + (lane & 15);

    const float* __restrict__ xrow = x + (size_t)m * D_IN + koff;
    const float* __restrict__ wcol = W + (size_t)koff * D_OUT + n;

    v8f acc = {0.f, 0.f, 0.f, 0.f, 0.f, 0.f, 0.f, 0.f};

#pragma unroll 4
    for (int k = 0; k < D_IN; k += 4) {
        v2f a = *(const v2f*)(xrow + k);                     // A[m][k+koff], A[m][k+koff+1]
        v2f b;
        b.x = wcol[(size_t)k * D_OUT];                       // B[k+koff][n]
        b.y = wcol[(size_t)k * D_OUT + D_OUT];               // B[k+koff+1][n]
        acc = __builtin_amdgcn_wmma_f32_16x16x4_f32(
                  /*neg_a=*/false, a, /*neg_b=*/false, b,
                  /*c_mod=*/(short)0, acc, /*reuse_a=*/false, /*reuse_b=*/false);
    }

    const int mr   = rbase + ((lane >> 4) << 3);             // row base for this half-wave
    const int ccol = (ct << 4) + (lane & 15);                // combined column 0..63
#pragma unroll
    for (int j = 0; j < 8; ++j)
        h[(size_t)(mr + j) * D2 + ccol] = acc[j];
}

// ---------------- out = b + dinv[i]^2 * h[i]  (self-loop + bias, inits all of d_out) ----

__global__ void k_init_out(const float* __restrict__ dinv, const float* __restrict__ h,
                           const float* __restrict__ bmu, const float* __restrict__ bls,
                           float* __restrict__ out) {
    long long idx = (long long)blockIdx.x * blockDim.x + threadIdx.x;
    if (idx >= (long long)N_NODES * D2) return;
    int i = (int)(idx >> 6);
    int j = (int)(idx & 63);
    float v   = dinv[i];
    float val = v * v * h[(size_t)i * D2 + j];
    if (j < D_OUT)
        out[(size_t)i * D_OUT + j] = val + bmu[j];
    else
        out[(size_t)N_NODES * D_OUT + (size_t)i * D_OUT + (j - D_OUT)] = val + bls[j - D_OUT];
}

// ---------------- edge scatter: one wave per edge, lane j -> cols j and j+32 ----------

__global__ __launch_bounds__(256) void k_scatter(const int* __restrict__ src,
                                                 const int* __restrict__ dst,
                                                 const float* __restrict__ dinv,
                                                 const float* __restrict__ h,
                                                 float* __restrict__ out) {
    long long t = (long long)blockIdx.x * blockDim.x + threadIdx.x;
    int e    = (int)(t >> 5);
    int lane = threadIdx.x & 31;
    if (e >= N_EDGES) return;
    int s = src[e];
    int d = dst[e];
    float w  = dinv[s] * dinv[d];
    float hm = h[(size_t)s * D2 + lane];
    float hl = h[(size_t)s * D2 + D_OUT + lane];
    unsafeAtomicAdd(&out[(size_t)d * D_OUT + lane], w * hm);
    unsafeAtomicAdd(&out[(size_t)N_NODES * D_OUT + (size_t)d * D_OUT + lane], w * hl);
}

// ---------------- launcher ----------------

extern "C" void kernel_launch(void* const* d_in, const int* in_sizes, int n_in,
                              void* d_out, int out_size, void* d_ws, size_t ws_size,
                              hipStream_t stream) {
    const float* x    = (const float*)d_in[0];
    const int*   ei   = (const int*)d_in[1];     // [2, E] flat; JAX default demotes int64->int32
    const float* Wmu  = (const float*)d_in[2];
    const float* bmu  = (const float*)d_in[3];
    const float* Wls  = (const float*)d_in[4];
    const float* bls  = (const float*)d_in[5];
    const int* src = ei;
    const int* dst = ei + N_EDGES;
    float* out = (float*)d_out;

    float* ws   = (float*)d_ws;
    float* dinv = ws;                 // N floats (deg -> dinv in place)
    float* h    = ws + (1 << 17);     // N*64 floats, 512KB-aligned offset

    k_deg_init <<<(N_NODES + 255) / 256, 256, 0, stream>>>(dinv);
    k_deg_count<<<(N_EDGES + 255) / 256, 256, 0, stream>>>(dst, dinv);
    k_dinv     <<<(N_NODES + 255) / 256, 256, 0, stream>>>(dinv);

    k_gemm<<<N_NODES / 32, 256, 0, stream>>>(x, Wmu, Wls, h);   // 100000 = 3125*32 exact

    long long outElems = (long long)N_NODES * D2;
    k_init_out<<<(unsigned)((outElems + 255) / 256), 256, 0, stream>>>(dinv, h, bmu, bls, out);

    long long scatThreads = (long long)N_EDGES * 32;
    k_scatter<<<(unsigned)((scatThreads + 255) / 256), 256, 0, stream>>>(src, dst, dinv, h, out);
}